// Mypnufft_grasp_test_57853209477811
// MI455X (gfx1250) — compile-verified
//
#include <hip/hip_runtime.h>
#include <math.h>

// Problem constants (match reference)
#define NN    256
#define NCOIL 8
#define NVEC  512
#define KTOT  (34 * 512)    // 17408
#define KB    32            // k-samples per block (two 16-row WMMA tiles)
#define NKB   (KTOT / KB)   // 544 (exact)

typedef __attribute__((ext_vector_type(2))) float v2f;
typedef __attribute__((ext_vector_type(8))) float v8f;

// ---------------------------------------------------------------------------
// Phase 1: Xc = C * x (complex), stored TRANSPOSED as planes XcT_r/XcT_i with
// layout [c][b][a] so WMMA B-fragment loads (lane = a) are coalesced.
// ---------------------------------------------------------------------------
__global__ __launch_bounds__(256) void xc_transpose_kernel(
    const float* __restrict__ input_r,  // (N,N,2)
    const float* __restrict__ C_r,      // (NC,N,N,2)
    float* __restrict__ xr,             // (NC,N,N)  [c][b][a]  real
    float* __restrict__ xi)             // (NC,N,N)  [c][b][a]  imag
{
  const int a = blockIdx.x & (NN - 1);
  const int c = blockIdx.x >> 8;
  const int b = threadIdx.x;
  const float x_re = input_r[(a * NN + b) * 2 + 0];
  const float x_im = input_r[(a * NN + b) * 2 + 1];
  const float c_re = C_r[((c * NN + a) * NN + b) * 2 + 0];
  const float c_im = C_r[((c * NN + a) * NN + b) * 2 + 1];
  xr[(c * NN + b) * NN + a] = c_re * x_re - c_im * x_im;
  xi[(c * NN + b) * NN + a] = c_re * x_im + c_im * x_re;
}

// ---------------------------------------------------------------------------
// Phase 2: per block = (32 k-samples = two 16-row WMMA k-tiles, coil c).
//   Stage A: build both E2 tiles exp(i*t_k*g_b) in LDS (64 KB, conflict-free).
//   Stage B: 8 waves x 2 a-tiles: complex GEMM over b=256 with
//            V_WMMA_F32_16X16X4_F32; each B fragment feeds BOTH k-tiles
//            (8 WMMAs per B load -> 2x arithmetic intensity vs 1 k-tile).
//   Stage C: multiply by E1[k,a]=exp(i*s_k*g_a) per lane, shfl-reduce over a,
//            cross-wave LDS reduce (aliased onto E2 space), *w, store.
// ---------------------------------------------------------------------------
__global__ __launch_bounds__(256) void nufft_wmma_kernel(
    const float* __restrict__ xr,     // [c][b][a]
    const float* __restrict__ xi,     // [c][b][a]
    const float* __restrict__ w,      // (NVEC)
    const float* __restrict__ angle,  // (K,2)  [:,0]=s (a-dim), [:,1]=t (b-dim)
    float* __restrict__ out)          // (NC,K,2)
{
  // Two E2 tiles, separate re/im planes, row stride 16 dwords:
  // lanes 0-15 hit banks (16b..16b+15), lanes 16-31 hit +32 dwords -> no conflicts.
  __shared__ float e2r[2][NN][16];    // 32 KB
  __shared__ float e2i[2][NN][16];    // 32 KB   (total exactly 64 KB)

  const int c   = blockIdx.y;
  const int k0  = blockIdx.x * KB;
  const int tid = threadIdx.x;
  const int lane = tid & 31;
  const int wave = tid >> 5;
  const int m16  = lane & 15;   // A: M(k-row); B/D: N(a-col)
  const int hi   = lane >> 4;   // upper half of wave

  // Lane l of every wave holds the angles of sample k0+l (shfl-broadcast later)
  const float s_reg = angle[(k0 + lane) * 2 + 0];
  const float t_reg = angle[(k0 + lane) * 2 + 1];

  // Stage A: generate both E2 tiles; thread tid owns column b = tid
  {
    const float gb = (float)tid - 128.0f;
    #pragma unroll
    for (int kt = 0; kt < 2; ++kt) {
      #pragma unroll
      for (int k = 0; k < 16; ++k) {
        const float t = __shfl(t_reg, kt * 16 + k, 32);
        float sn, cs;
        __sincosf(t * gb, &sn, &cs);
        e2r[kt][tid][k] = cs;
        e2i[kt][tid][k] = sn;
      }
    }
  }
  __syncthreads();

  const float* __restrict__ xrp = xr + c * NN * NN;
  const float* __restrict__ xip = xi + c * NN * NN;

  float pr[2][8], pi[2][8];
  #pragma unroll
  for (int kt = 0; kt < 2; ++kt)
    #pragma unroll
    for (int j = 0; j < 8; ++j) { pr[kt][j] = 0.0f; pi[kt][j] = 0.0f; }

  // Stage B: each wave handles 2 of the 16 a-tiles; both k-tiles per B load
  for (int t = 0; t < 2; ++t) {
    const int a0   = (wave * 2 + t) * 16;
    const int acol = a0 + m16;
    v8f accr0 = {}, acci0 = {};   // k-tile 0
    v8f accr1 = {}, acci1 = {};   // k-tile 1

    #pragma unroll 4
    for (int b0 = 0; b0 < NN; b0 += 4) {
      const int bb = b0 + hi * 2;
      // B fragments (XcT) from global: lane holds (K=bb-b0, N=acol) — coalesced
      v2f br, bi;
      br.x = xrp[bb * NN + acol];      br.y = xrp[(bb + 1) * NN + acol];
      bi.x = xip[bb * NN + acol];      bi.y = xip[(bb + 1) * NN + acol];
      // A fragments for both k-tiles from LDS
      v2f ar0, ai0, na0, ar1, ai1, na1;
      ar0.x = e2r[0][bb][m16];  ar0.y = e2r[0][bb + 1][m16];
      ai0.x = e2i[0][bb][m16];  ai0.y = e2i[0][bb + 1][m16];
      ar1.x = e2r[1][bb][m16];  ar1.y = e2r[1][bb + 1][m16];
      ai1.x = e2i[1][bb][m16];  ai1.y = e2i[1][bb + 1][m16];
      na0.x = -ai0.x;  na0.y = -ai0.y;
      na1.x = -ai1.x;  na1.y = -ai1.y;
      // Complex accumulate: Re += E2r*Xr - E2i*Xi ; Im += E2r*Xi + E2i*Xr
      accr0 = __builtin_amdgcn_wmma_f32_16x16x4_f32(false, ar0, false, br,
                                                    (short)0, accr0, false, false);
      accr0 = __builtin_amdgcn_wmma_f32_16x16x4_f32(false, na0, false, bi,
                                                    (short)0, accr0, false, false);
      acci0 = __builtin_amdgcn_wmma_f32_16x16x4_f32(false, ar0, false, bi,
                                                    (short)0, acci0, false, false);
      acci0 = __builtin_amdgcn_wmma_f32_16x16x4_f32(false, ai0, false, br,
                                                    (short)0, acci0, false, false);
      accr1 = __builtin_amdgcn_wmma_f32_16x16x4_f32(false, ar1, false, br,
                                                    (short)0, accr1, false, false);
      accr1 = __builtin_amdgcn_wmma_f32_16x16x4_f32(false, na1, false, bi,
                                                    (short)0, accr1, false, false);
      acci1 = __builtin_amdgcn_wmma_f32_16x16x4_f32(false, ar1, false, bi,
                                                    (short)0, acci1, false, false);
      acci1 = __builtin_amdgcn_wmma_f32_16x16x4_f32(false, ai1, false, br,
                                                    (short)0, acci1, false, false);
    }

    // Stage C: fold in E1[k,a] = exp(i*s_k*g_a); D layout: vgpr j = M row (+8 hi)
    const float ga = (float)acol - 128.0f;
    #pragma unroll
    for (int kt = 0; kt < 2; ++kt) {
      #pragma unroll
      for (int j = 0; j < 8; ++j) {
        const int kl = kt * 16 + j + hi * 8;
        const float s = __shfl(s_reg, kl, 32);
        float sn, cs;
        __sincosf(s * ga, &sn, &cs);
        const float mr = (kt == 0) ? accr0[j] : accr1[j];
        const float mi = (kt == 0) ? acci0[j] : acci1[j];
        pr[kt][j] += mr * cs - mi * sn;
        pi[kt][j] += mr * sn + mi * cs;
      }
    }
  }

  // Reduce over a within each 16-lane half; keep results in registers
  float outr[2][8], outq[2][8];
  #pragma unroll
  for (int kt = 0; kt < 2; ++kt) {
    #pragma unroll
    for (int j = 0; j < 8; ++j) {
      float r = pr[kt][j], q = pi[kt][j];
      #pragma unroll
      for (int off = 1; off < 16; off <<= 1) {
        r += __shfl_xor(r, off, 32);
        q += __shfl_xor(q, off, 32);
      }
      outr[kt][j] = r;
      outq[kt][j] = q;
    }
  }

  __syncthreads();                        // all waves done reading E2 tiles
  float* yred = &e2r[0][0][0];            // alias: [wave][32 k][2] = 2 KB

  if (m16 == 0) {                         // lanes 0 and 16 hold the half-sums
    #pragma unroll
    for (int kt = 0; kt < 2; ++kt) {
      #pragma unroll
      for (int j = 0; j < 8; ++j) {
        const int kl = kt * 16 + j + hi * 8;
        yred[(wave * KB + kl) * 2 + 0] = outr[kt][j];
        yred[(wave * KB + kl) * 2 + 1] = outq[kt][j];
      }
    }
  }
  __syncthreads();

  // Cross-wave reduce, apply w, store
  if (tid < 2 * KB) {
    const int kl = tid >> 1, comp = tid & 1;
    float s = 0.0f;
    #pragma unroll
    for (int wv = 0; wv < 8; ++wv) s += yred[(wv * KB + kl) * 2 + comp];
    const int kk = k0 + kl;
    s *= w[kk & (NVEC - 1)];
    out[(c * KTOT + kk) * 2 + comp] = s;
  }
}

// ---------------------------------------------------------------------------
extern "C" void kernel_launch(void* const* d_in, const int* in_sizes, int n_in,
                              void* d_out, int out_size, void* d_ws, size_t ws_size,
                              hipStream_t stream) {
  const float* input_r = (const float*)d_in[0];  // (256,256,2)
  const float* C_r     = (const float*)d_in[1];  // (8,256,256,2)
  const float* w       = (const float*)d_in[2];  // (512,)
  const float* angle   = (const float*)d_in[3];  // (17408,2)
  float* out = (float*)d_out;

  // Workspace: XcT real plane then imag plane, each NC*N*N floats (4 MB total)
  float* xr = (float*)d_ws;
  float* xi = xr + NCOIL * NN * NN;

  xc_transpose_kernel<<<NCOIL * NN, 256, 0, stream>>>(input_r, C_r, xr, xi);

  dim3 grid(NKB, NCOIL);
  nufft_wmma_kernel<<<grid, 256, 0, stream>>>(xr, xi, w, angle, out);
}